// MambaBackbone_56642028700212
// MI455X (gfx1250) — compile-verified
//
#include <hip/hip_runtime.h>
#include <math.h>

// Model constants (from reference)
#define L_SEQ   1024
#define IN_LEN_ 512
#define DMODEL  1024
#define DINNER  2048
#define DSTATE  16
#define DTRANK  64
#define XDBL    96       // DT_RANK + 2*D_STATE
#define NLAYERS 4
#define LN_EPS  1e-5f

typedef __attribute__((ext_vector_type(2))) float v2f;
typedef __attribute__((ext_vector_type(8))) float v8f;

// ---------------------------------------------------------------------------
// WMMA fp32 GEMM:  C[M,N] = A[M,K] @ B[N,K]^T  (+bias, +optional softplus)
// A row-major (lda), B row-major N x K (ldb).  One wave computes a
// 32 x (16*JT) tile via 2 x JT fragments of V_WMMA_F32_16X16X4_F32.
// JT=4 -> 32x64 tile (8 WMMAs per 6 b64 loads per k-step); JT=2 -> 32x32.
// ---------------------------------------------------------------------------
template <int JT>
__global__ __launch_bounds__(256) void wmma_gemm_f32(
    const float* __restrict__ A, int lda,
    const float* __restrict__ Bm, int ldb,
    float* __restrict__ C, int ldc,
    int M, int N, int K,
    const float* __restrict__ bias, int act)
{
    const int tilesN = N / (16 * JT);
    const int tilesM = M >> 5;
    const int wid = blockIdx.x * (blockDim.x >> 5) + (threadIdx.x >> 5);
    if (wid >= tilesM * tilesN) return;   // wave-uniform: EXEC stays all-1s

    const int tm = wid / tilesN;
    const int tn = wid - tm * tilesN;
    const int m0 = tm << 5;
    const int n0 = tn * (16 * JT);

    const int lane = threadIdx.x & 31;
    const int half = lane >> 4;           // selects K pair base (0 or 2)
    const int r    = lane & 15;           // row (A) / col (B) within 16-tile

    const float* ap[2];
    ap[0] = A + (size_t)(m0      + r) * lda;
    ap[1] = A + (size_t)(m0 + 16 + r) * lda;
    const float* bp[JT];
    #pragma unroll
    for (int j = 0; j < JT; ++j)
        bp[j] = Bm + (size_t)(n0 + 16 * j + r) * ldb;

    v8f c[2][JT] = {};

    for (int k0 = 0; k0 < K; k0 += 4) {
        const int kk = k0 + 2 * half;     // even -> 8B-aligned float2 loads
        v2f a[2], b[JT];
        #pragma unroll
        for (int i = 0; i < 2; ++i)  a[i] = *(const v2f*)(ap[i] + kk);
        #pragma unroll
        for (int j = 0; j < JT; ++j) b[j] = *(const v2f*)(bp[j] + kk);
        #pragma unroll
        for (int i = 0; i < 2; ++i) {
            #pragma unroll
            for (int j = 0; j < JT; ++j) {
                // 8 args: (neg_a, A, neg_b, B, c_mod, C, reuse_a, reuse_b)
                c[i][j] = __builtin_amdgcn_wmma_f32_16x16x4_f32(
                    false, a[i], false, b[j], (short)0, c[i][j], false, false);
            }
        }
    }

    #pragma unroll
    for (int i = 0; i < 2; ++i) {
        #pragma unroll
        for (int j = 0; j < JT; ++j) {
            const int col = n0 + 16 * j + r;
            const float bv = bias ? bias[col] : 0.0f;
            #pragma unroll
            for (int v = 0; v < 8; ++v) {
                const int row = m0 + 16 * i + 8 * half + v;   // C layout: M = v + 8*half
                float val = c[i][j][v] + bv;
                if (act == 1) val = (val > 20.0f) ? val : log1pf(expf(val));  // softplus
                C[(size_t)row * ldc + col] = val;
            }
        }
    }
}

// ---------------------------------------------------------------------------
// Residual add + LayerNorm, one 256-thread block per row (D_MODEL = 1024).
// ---------------------------------------------------------------------------
__global__ __launch_bounds__(256) void resnorm_kernel(
    const float* __restrict__ h, float* __restrict__ res, float* __restrict__ hn,
    const float* __restrict__ w, const float* __restrict__ b, int first)
{
    const int row = blockIdx.x;
    const int tid = threadIdx.x;
    __shared__ float partial[8];
    __shared__ float statS;

    float vals[4];
    float lsum = 0.0f;
    #pragma unroll
    for (int i = 0; i < 4; ++i) {
        const int col = tid + (i << 8);
        const size_t idx = (size_t)row * DMODEL + col;
        float v = h[idx];
        if (!first) v += res[idx];
        res[idx] = v;
        vals[i] = v;
        lsum += v;
    }
    for (int o = 16; o > 0; o >>= 1) lsum += __shfl_xor(lsum, o, 32);
    if ((tid & 31) == 0) partial[tid >> 5] = lsum;
    __syncthreads();
    if (tid == 0) {
        float t = 0.0f;
        for (int i = 0; i < 8; ++i) t += partial[i];
        statS = t * (1.0f / DMODEL);
    }
    __syncthreads();
    const float mu = statS;

    float lsq = 0.0f;
    #pragma unroll
    for (int i = 0; i < 4; ++i) { const float cdiff = vals[i] - mu; lsq += cdiff * cdiff; }
    __syncthreads();   // everyone has read mu before statS/partial reuse
    for (int o = 16; o > 0; o >>= 1) lsq += __shfl_xor(lsq, o, 32);
    if ((tid & 31) == 0) partial[tid >> 5] = lsq;
    __syncthreads();
    if (tid == 0) {
        float t = 0.0f;
        for (int i = 0; i < 8; ++i) t += partial[i];
        statS = t * (1.0f / DMODEL);
    }
    __syncthreads();
    const float rstd = rsqrtf(statS + LN_EPS);

    #pragma unroll
    for (int i = 0; i < 4; ++i) {
        const int col = tid + (i << 8);
        hn[(size_t)row * DMODEL + col] = (vals[i] - mu) * rstd * w[col] + b[col];
    }
}

// ---------------------------------------------------------------------------
// Depthwise causal conv (width 4) + bias + SiLU.  x = xz[:, :DINNER].
// ---------------------------------------------------------------------------
__global__ __launch_bounds__(256) void conv_silu_kernel(
    const float* __restrict__ xz, const float* __restrict__ cw,
    const float* __restrict__ cb, float* __restrict__ xc)
{
    const int idx = blockIdx.x * blockDim.x + threadIdx.x;
    if (idx >= L_SEQ * DINNER) return;
    const int l = idx >> 11;          // / DINNER
    const int d = idx & (DINNER - 1);
    float acc = cb[d];
    #pragma unroll
    for (int k = 0; k < 4; ++k) {
        const int ls = l - 3 + k;
        if (ls >= 0) acc += cw[d * 4 + k] * xz[(size_t)ls * (2 * DINNER) + d];
    }
    xc[idx] = acc / (1.0f + expf(-acc));   // silu
}

// ---------------------------------------------------------------------------
// Selective scan, state-parallel: one thread per (channel, state) pair
// (32768 threads = 1024 waves vs 64 waves for channel-parallel).  The 16
// states of a channel occupy 16 consecutive lanes of a wave half; y is
// produced with a 4-step __shfl_xor reduction (stays within each 16-lane
// group).  Gating  y = (sum + x*D) * silu(z)  is fused into the epilogue.
// Serial dependence per step is a single FMA (h = e*h + f; e,f load-derived),
// so unrolling lets loads for future timesteps issue ahead.
// ---------------------------------------------------------------------------
__global__ __launch_bounds__(256) void scan_state_kernel(
    const float* __restrict__ delta, const float* __restrict__ xc,
    const float* __restrict__ x_dbl, const float* __restrict__ A_log,
    const float* __restrict__ xz, const float* __restrict__ ssm_D,
    float* __restrict__ ys)
{
    const int p = blockIdx.x * blockDim.x + threadIdx.x;   // DINNER*DSTATE
    const int d = p >> 4;
    const int s = p & (DSTATE - 1);

    const float A  = -expf(A_log[(size_t)d * DSTATE + s]);
    const float Dv = ssm_D[d];
    float h = 0.0f;

    #pragma unroll 4
    for (int t = 0; t < L_SEQ; ++t) {
        const float dt = delta[(size_t)t * DINNER + d];          // broadcast x16
        const float xt = xc[(size_t)t * DINNER + d];             // broadcast x16
        const float Bv = x_dbl[(size_t)t * XDBL + DTRANK + s];
        const float Cv = x_dbl[(size_t)t * XDBL + DTRANK + DSTATE + s];

        h = expf(dt * A) * h + (dt * Bv) * xt;

        float contrib = h * Cv;
        contrib += __shfl_xor(contrib, 1, 32);
        contrib += __shfl_xor(contrib, 2, 32);
        contrib += __shfl_xor(contrib, 4, 32);
        contrib += __shfl_xor(contrib, 8, 32);
        if (s == 0) {
            const float z = xz[(size_t)t * (2 * DINNER) + DINNER + d];
            const float y = contrib + xt * Dv;
            ys[(size_t)t * DINNER + d] = y * (z / (1.0f + expf(-z)));
        }
    }
}

// ---------------------------------------------------------------------------
// Host orchestration
// ---------------------------------------------------------------------------
extern "C" void kernel_launch(void* const* d_in, const int* in_sizes, int n_in,
                              void* d_out, int out_size, void* d_ws, size_t ws_size,
                              hipStream_t stream)
{
    (void)in_sizes; (void)n_in; (void)out_size; (void)ws_size;

    const float* input     = (const float*)d_in[0];
    // d_in[1] = inference_parameters (unused)
    const float* input_w   = (const float*)d_in[2];
    const float* norm_w    = (const float*)d_in[3];
    const float* norm_b    = (const float*)d_in[4];
    const float* in_proj_w = (const float*)d_in[5];
    const float* conv_w    = (const float*)d_in[6];
    const float* conv_b    = (const float*)d_in[7];
    const float* x_proj_w  = (const float*)d_in[8];
    const float* dt_proj_w = (const float*)d_in[9];
    const float* dt_proj_b = (const float*)d_in[10];
    const float* A_log     = (const float*)d_in[11];
    const float* ssm_D     = (const float*)d_in[12];
    const float* out_proj_w= (const float*)d_in[13];

    float* ws = (float*)d_ws;
    size_t off = 0;
    float* h     = ws + off;  off += (size_t)L_SEQ * DMODEL;       // 1M
    float* res   = ws + off;  off += (size_t)L_SEQ * DMODEL;       // 1M
    float* hn    = ws + off;  off += (size_t)L_SEQ * DMODEL;       // 1M
    float* xz    = ws + off;  off += (size_t)L_SEQ * 2 * DINNER;   // 4M
    float* xc    = ws + off;  off += (size_t)L_SEQ * DINNER;       // 2M
    float* x_dbl = ws + off;  off += (size_t)L_SEQ * XDBL;         // 96K
    float* delta = ws + off;  off += (size_t)L_SEQ * DINNER;       // 2M
    float* ys    = ws + off;  off += (size_t)L_SEQ * DINNER;       // 2M

    // JT=4 -> 32x64 wave tile; JT=2 -> 32x32 (for N=96)
    auto gemm64 = [&](const float* A, int lda, const float* B, int ldb,
                      float* C, int ldc, int M, int N, int K,
                      const float* bias, int act) {
        const int tiles  = (M >> 5) * (N >> 6);
        const int blocks = (tiles + 7) / 8;   // 8 waves / 256-thread block
        wmma_gemm_f32<4><<<blocks, 256, 0, stream>>>(A, lda, B, ldb, C, ldc,
                                                     M, N, K, bias, act);
    };
    auto gemm32 = [&](const float* A, int lda, const float* B, int ldb,
                      float* C, int ldc, int M, int N, int K,
                      const float* bias, int act) {
        const int tiles  = (M >> 5) * (N >> 5);
        const int blocks = (tiles + 7) / 8;
        wmma_gemm_f32<2><<<blocks, 256, 0, stream>>>(A, lda, B, ldb, C, ldc,
                                                     M, N, K, bias, act);
    };

    // h = input @ input_w^T        (1024x512) x (1024x512)^T -> (1024x1024)
    gemm64(input, IN_LEN_, input_w, IN_LEN_, h, DMODEL, L_SEQ, DMODEL, IN_LEN_, nullptr, 0);

    const int elemBlocks = (L_SEQ * DINNER) / 256;

    for (int i = 0; i < NLAYERS; ++i) {
        // residual + layernorm
        resnorm_kernel<<<L_SEQ, 256, 0, stream>>>(
            h, res, hn, norm_w + (size_t)i * DMODEL, norm_b + (size_t)i * DMODEL, i == 0);

        // xz = hn @ in_proj_w^T    (1024x1024) x (4096x1024)^T -> (1024x4096)
        gemm64(hn, DMODEL, in_proj_w + (size_t)i * 2 * DINNER * DMODEL, DMODEL,
               xz, 2 * DINNER, L_SEQ, 2 * DINNER, DMODEL, nullptr, 0);

        // depthwise conv + silu -> xc
        conv_silu_kernel<<<elemBlocks, 256, 0, stream>>>(
            xz, conv_w + (size_t)i * DINNER * 4, conv_b + (size_t)i * DINNER, xc);

        // x_dbl = xc @ x_proj_w^T  (1024x2048) x (96x2048)^T -> (1024x96)
        gemm32(xc, DINNER, x_proj_w + (size_t)i * XDBL * DINNER, DINNER,
               x_dbl, XDBL, L_SEQ, XDBL, DINNER, nullptr, 0);

        // delta = softplus(x_dbl[:, :64] @ dt_proj_w^T + dt_bias) -> (1024x2048)
        gemm64(x_dbl, XDBL, dt_proj_w + (size_t)i * DINNER * DTRANK, DTRANK,
               delta, DINNER, L_SEQ, DINNER, DTRANK, dt_proj_b + (size_t)i * DINNER, 1);

        // selective scan (state-parallel) + fused gating -> ys
        scan_state_kernel<<<(DINNER * DSTATE) / 256, 256, 0, stream>>>(
            delta, xc, x_dbl, A_log + (size_t)i * DINNER * DSTATE,
            xz, ssm_D + (size_t)i * DINNER, ys);

        // out = ys @ out_proj_w^T  (1024x2048) x (1024x2048)^T -> (1024x1024)
        float* outp = (i == NLAYERS - 1) ? (float*)d_out : h;
        gemm64(ys, DINNER, out_proj_w + (size_t)i * DMODEL * DINNER, DINNER,
               outp, DMODEL, L_SEQ, DMODEL, DINNER, nullptr, 0);
    }
}